// CBOW_38027640439069
// MI455X (gfx1250) — compile-verified
//
#include <hip/hip_runtime.h>
#include <math.h>

typedef __attribute__((ext_vector_type(2))) float v2f;
typedef __attribute__((ext_vector_type(8))) float v8f;

#define WAVES_PER_BLOCK 4
#define ROWS_PER_WAVE   16
#define LDS_STRIDE      132   // padded row stride (floats): conflict-free phase-2 pattern
#define EMBED           128
#define CTX             8
#define NEGS            5
#define NT              6     // 1 positive + 5 negative targets

// ---------------------------------------------------------------------------
// Kernel 1: one wave handles a 16-row batch tile.
//   Phase 1: Ubar[row] = sum of 8 context embeddings -> LDS (padded stride).
//   Phase 2: for each of 6 targets, 32 chunks of V_WMMA_F32_16X16X4_F32
//            accumulate diag(Ubar_tile * T_tile^T) = per-row dot products.
// ---------------------------------------------------------------------------
__global__ __launch_bounds__(WAVES_PER_BLOCK * 32)
void cbow_partials_kernel(const int* __restrict__ pos_u,
                          const int* __restrict__ pos_w,
                          const int* __restrict__ neg_w,
                          const float* __restrict__ W,
                          float* __restrict__ partials)
{
    __shared__ float lds[WAVES_PER_BLOCK * ROWS_PER_WAVE * LDS_STRIDE];

    const int lane    = threadIdx.x & 31;
    const int wib     = threadIdx.x >> 5;
    const int tile    = blockIdx.x * WAVES_PER_BLOCK + wib;
    const int rowBase = tile * ROWS_PER_WAVE;
    float* my = lds + wib * (ROWS_PER_WAVE * LDS_STRIDE);

    // ---- Phase 1: coalesced full-row gathers (512B per row), sum contexts ----
    for (int rr = 0; rr < ROWS_PER_WAVE; ++rr) {
        const int row = rowBase + rr;
        const int* pu = pos_u + row * CTX;
        float ax = 0.f, ay = 0.f, az = 0.f, aw = 0.f;
#pragma unroll
        for (int c = 0; c < CTX; ++c) {
            const float4 v =
                *(const float4*)(W + (size_t)pu[c] * EMBED + lane * 4);
            ax += v.x; ay += v.y; az += v.z; aw += v.w;
        }
        float* dst = my + rr * LDS_STRIDE + lane * 4;   // 8B aligned (stride 132)
        ((float2*)dst)[0] = make_float2(ax, ay);
        ((float2*)dst)[1] = make_float2(az, aw);
    }
    __syncthreads();

    // ---- Phase 2: WMMA dot products ----
    const int rl = lane & 15;   // row within tile (also N column for this lane)
    const int h  = lane >> 4;   // lane half selects K sub-pair {0,1} vs {2,3}

    for (int t = 0; t < NT; ++t) {
        const int tIdx = (t == 0)
            ? pos_w[rowBase + rl]
            : neg_w[(rowBase + rl) * NEGS + (t - 1)];
        const float* trow = W + (size_t)tIdx * EMBED;
        const float* arow = my + rl * LDS_STRIDE;

        v8f c8 = {0.f, 0.f, 0.f, 0.f, 0.f, 0.f, 0.f, 0.f};
#pragma unroll 8
        for (int i = 0; i < 32; ++i) {
            const int off = 4 * i + 2 * h;
            // A[m,k]=Ubar[m][4i+k] and B[k,n]=T[n][4i+k] share the same
            // per-lane layout: lane L holds elems {off, off+1} of row L&15.
            v2f a = *(const v2f*)(arow + off);   // ds_load_b64, bank-conflict-free
            v2f b = *(const v2f*)(trow + off);   // global_load_b64, L2-resident
            c8 = __builtin_amdgcn_wmma_f32_16x16x4_f32(
                     false, a, false, b, (short)0, c8, false, false);
        }

        // diag(D): (j,j) j<8 -> VGPR j lane j ; j>=8 -> VGPR j-8 lane j+16
        const int sel = (lane < 8) ? lane : ((lane >= 24) ? (lane - 24) : -1);
        float d = c8[0];
        d = (sel == 1) ? c8[1] : d;
        d = (sel == 2) ? c8[2] : d;
        d = (sel == 3) ? c8[3] : d;
        d = (sel == 4) ? c8[4] : d;
        d = (sel == 5) ? c8[5] : d;
        d = (sel == 6) ? c8[6] : d;
        d = (sel == 7) ? c8[7] : d;
        d = (sel < 0) ? 0.f : d;

        // wave32 reduction of the 16 diagonal contributions
#pragma unroll
        for (int o = 16; o >= 1; o >>= 1)
            d += __shfl_xor(d, o, 32);

        if (lane == 0)
            partials[tile * NT + t] = d;
    }
}

// ---------------------------------------------------------------------------
// Kernel 2: deterministic single-block reduction + loss epilogue.
// ---------------------------------------------------------------------------
__device__ __forceinline__ float softplus_f(float x)
{
    // -log_sigmoid(-x) = log(1 + e^x), guarded for large x
    return (x > 20.f) ? x : log1pf(expf(x));
}

__global__ __launch_bounds__(256)
void cbow_finalize_kernel(const float* __restrict__ partials,
                          float* __restrict__ out, int ntiles)
{
    __shared__ float red[NT][256];
    const int t = threadIdx.x;

    float s[NT];
#pragma unroll
    for (int n = 0; n < NT; ++n) s[n] = 0.f;

    for (int i = t; i < ntiles; i += 256) {
#pragma unroll
        for (int n = 0; n < NT; ++n) s[n] += partials[i * NT + n];
    }
#pragma unroll
    for (int n = 0; n < NT; ++n) red[n][t] = s[n];
    __syncthreads();

    for (int stride = 128; stride >= 1; stride >>= 1) {
        if (t < stride) {
#pragma unroll
            for (int n = 0; n < NT; ++n) red[n][t] += red[n][t + stride];
        }
        __syncthreads();
    }

    if (t == 0) {
        float loss = softplus_f(-red[0][0]);        // -log_sigmoid(s_pos)
#pragma unroll
        for (int n = 1; n < NT; ++n)
            loss += softplus_f(red[n][0]);          // -log_sigmoid(-s_neg)
        out[0] = loss;
    }
}

// ---------------------------------------------------------------------------
extern "C" void kernel_launch(void* const* d_in, const int* in_sizes, int n_in,
                              void* d_out, int out_size, void* d_ws, size_t ws_size,
                              hipStream_t stream)
{
    const int*   pos_u = (const int*)d_in[0];   // [B, 8]
    const int*   pos_w = (const int*)d_in[1];   // [B]
    const int*   neg_w = (const int*)d_in[2];   // [B, 5]
    const float* W     = (const float*)d_in[3]; // [VOCAB, 128]
    float*       out   = (float*)d_out;
    float*       partials = (float*)d_ws;       // [tiles * 6] floats

    const int batch  = in_sizes[1];             // 16384
    const int tiles  = batch / ROWS_PER_WAVE;   // 1024 (batch divisible by 16)
    const int blocks = tiles / WAVES_PER_BLOCK; // 256

    cbow_partials_kernel<<<blocks, WAVES_PER_BLOCK * 32, 0, stream>>>(
        pos_u, pos_w, neg_w, W, partials);
    cbow_finalize_kernel<<<1, 256, 0, stream>>>(partials, out, tiles);
}